// ImageCrossMultiply_31069793419620
// MI455X (gfx1250) — compile-verified
//
#include <hip/hip_runtime.h>

// Problem constants (from reference): B=8, F=64, S=256
constexpr int NB   = 8;
constexpr int NF   = 64;
constexpr int NS   = 256;
constexpr int NPIX = NS * NS;      // 65536
constexpr int KCHUNKS = 32;        // K-split for the cross GEMM

typedef __attribute__((ext_vector_type(2))) float v2f;
typedef __attribute__((ext_vector_type(4))) float v4f;
typedef __attribute__((ext_vector_type(8))) float v8f;

// ---------------------------------------------------------------------------
// Kernel 0: zero the mults accumulation region (atomically accumulated later)
// ---------------------------------------------------------------------------
__global__ void zero_mults_kernel(float* __restrict__ out) {
    out[blockIdx.x * blockDim.x + threadIdx.x] = 0.0f;
}

// ---------------------------------------------------------------------------
// Kernel 1: affine grid + bilinear grid_sample (zeros padding, align_corners=F)
// One thread per (b, h, w) pixel; bilinear corner offsets/weights computed
// once and reused across all 64 features (coalesced loads & stores per f).
// ---------------------------------------------------------------------------
__global__ void __launch_bounds__(256)
warp_sample_kernel(const float* __restrict__ vol2,
                   const float* __restrict__ alpha,
                   const float* __restrict__ dr,
                   float* __restrict__ out_v2w,
                   float* __restrict__ out_grid) {
    const int p  = blockIdx.x * blockDim.x + threadIdx.x;   // 0 .. NB*NPIX
    const int b  = p >> 16;                                 // NPIX == 65536
    const int hw = p & (NPIX - 1);
    const int h  = hw >> 8;
    const int w  = hw & (NS - 1);

    // Per-batch affine matrix A = [[c, -s, t0], [s, c, t1]]
    const float al = alpha[b];
    float sn, cs;
    sincosf(al, &sn, &cs);
    const float inv_s = 1.0f / (float)NS;
    const float drn0 = -2.0f * dr[2 * b + 1] * inv_s;       // dr reversed, scaled
    const float drn1 = -2.0f * dr[2 * b + 0] * inv_s;
    const float t0 = cs * drn0 - sn * drn1;
    const float t1 = sn * drn0 + cs * drn1;

    // Normalized pixel-center coordinates
    const float xs = (2.0f * (float)w + 1.0f) * inv_s - 1.0f;
    const float ys = (2.0f * (float)h + 1.0f) * inv_s - 1.0f;
    const float gx = cs * xs - sn * ys + t0;
    const float gy = sn * xs + cs * ys + t1;

    // grid output [B,S,S,2] -> flat index p*2 (8-byte vector store)
    v2f g; g.x = gx; g.y = gy;
    *(v2f*)(out_grid + (size_t)p * 2) = g;

    // Un-normalize (align_corners=False)
    const float ixf = ((gx + 1.0f) * (float)NS - 1.0f) * 0.5f;
    const float iyf = ((gy + 1.0f) * (float)NS - 1.0f) * 0.5f;
    const float x0f = floorf(ixf), y0f = floorf(iyf);
    const float fx = ixf - x0f, fy = iyf - y0f;
    const int ix0 = (int)x0f, iy0 = (int)y0f;
    const int ix1 = ix0 + 1,  iy1 = iy0 + 1;

    // zeros padding: fold validity into weights, clamp indices for the gather
    const float vx0 = (ix0 >= 0 && ix0 < NS) ? 1.0f : 0.0f;
    const float vx1 = (ix1 >= 0 && ix1 < NS) ? 1.0f : 0.0f;
    const float vy0 = (iy0 >= 0 && iy0 < NS) ? 1.0f : 0.0f;
    const float vy1 = (iy1 >= 0 && iy1 < NS) ? 1.0f : 0.0f;
    const int cx0 = min(max(ix0, 0), NS - 1);
    const int cx1 = min(max(ix1, 0), NS - 1);
    const int cy0 = min(max(iy0, 0), NS - 1);
    const int cy1 = min(max(iy1, 0), NS - 1);

    const float w00 = (1.0f - fy) * (1.0f - fx) * vy0 * vx0;
    const float w01 = (1.0f - fy) * fx          * vy0 * vx1;
    const float w10 = fy          * (1.0f - fx) * vy1 * vx0;
    const float w11 = fy          * fx          * vy1 * vx1;

    const int o00 = cy0 * NS + cx0;
    const int o01 = cy0 * NS + cx1;
    const int o10 = cy1 * NS + cx0;
    const int o11 = cy1 * NS + cx1;

    const float* __restrict__ base2 = vol2 + (size_t)b * NF * NPIX;
    float* __restrict__ outp = out_v2w + (size_t)b * NF * NPIX + hw;

#pragma unroll 4
    for (int f = 0; f < NF; ++f) {
        const float* __restrict__ pf = base2 + (size_t)f * NPIX;
        const float v = pf[o00] * w00 + pf[o01] * w01 +
                        pf[o10] * w10 + pf[o11] * w11;
        outp[(size_t)f * NPIX] = v;
    }
}

// ---------------------------------------------------------------------------
// Kernel 2: mults[b,i,j] = sum_p v1[b,i,p] * v2w[b,j,p]
// Per-batch (64 x 65536) x (65536 x 64) GEMM via V_WMMA_F32_16X16X4_F32.
// Block = 512 threads = 16 waves; wave wv owns output tile (wv/4, wv%4) of the
// 4x4 grid of 16x16 tiles. K split across KCHUNKS blocks per batch; partial
// tiles accumulated into d_out with global_atomic_add_f32.
//
// A fragment (16x4 f32): lane l holds row M=l%16, k-slot = (l/16)*2 + vgpr.
// The k-slot -> pixel mapping is arbitrary (pure dot product) as long as A and
// B agree, so each lane reads two b128 chunks per 16-pixel step and feeds 4
// WMMAs from register pieces.
// ---------------------------------------------------------------------------
__global__ void __launch_bounds__(512)
cross_gemm_kernel(const float* __restrict__ v1,
                  const float* __restrict__ v2w,
                  float* __restrict__ out_mults) {
    const int b    = blockIdx.x / KCHUNKS;
    const int kc   = blockIdx.x % KCHUNKS;
    const int wv   = threadIdx.x >> 5;     // wave 0..15
    const int lane = threadIdx.x & 31;
    const int half = lane >> 4;            // 0 | 1
    const int lm   = lane & 15;
    const int tm   = (wv >> 2) << 4;       // tile row base: 0,16,32,48
    const int tn   = (wv & 3) << 4;        // tile col base

    const float* __restrict__ arow = v1  + ((size_t)b * NF + tm + lm) * NPIX;
    const float* __restrict__ brow = v2w + ((size_t)b * NF + tn + lm) * NPIX;

    constexpr int KLEN = NPIX / KCHUNKS;   // 2048 pixels per block
    const int k0 = kc * KLEN;
    const int k1 = k0 + KLEN;

    v8f acc = {0.f, 0.f, 0.f, 0.f, 0.f, 0.f, 0.f, 0.f};

    for (int k = k0; k < k1; k += 16) {
        const v4f a1 = *(const v4f*)(arow + k +     half * 8);
        const v4f a2 = *(const v4f*)(arow + k + 4 + half * 8);
        const v4f b1 = *(const v4f*)(brow + k +     half * 8);
        const v4f b2 = *(const v4f*)(brow + k + 4 + half * 8);
        acc = __builtin_amdgcn_wmma_f32_16x16x4_f32(false, a1.xy, false, b1.xy,
                                                    (short)0, acc, false, false);
        acc = __builtin_amdgcn_wmma_f32_16x16x4_f32(false, a1.zw, false, b1.zw,
                                                    (short)0, acc, false, false);
        acc = __builtin_amdgcn_wmma_f32_16x16x4_f32(false, a2.xy, false, b2.xy,
                                                    (short)0, acc, false, false);
        acc = __builtin_amdgcn_wmma_f32_16x16x4_f32(false, a2.zw, false, b2.zw,
                                                    (short)0, acc, false, false);
    }

    // C/D layout: vgpr r, lanes 0-15 -> M=r; lanes 16-31 -> M=8+r; N=lane%16
    float* __restrict__ outt =
        out_mults + (size_t)b * NF * NF + (size_t)(tm + half * 8) * NF + tn + lm;
#pragma unroll
    for (int r = 0; r < 8; ++r) {
        unsafeAtomicAdd(outt + (size_t)r * NF, acc[r]);
    }
}

// ---------------------------------------------------------------------------
// Launch: d_out = [ mults (B*F*F) | v2w (B*F*S*S) | grid (B*S*S*2) ]
// ---------------------------------------------------------------------------
extern "C" void kernel_launch(void* const* d_in, const int* in_sizes, int n_in,
                              void* d_out, int out_size, void* d_ws, size_t ws_size,
                              hipStream_t stream) {
    const float* volume1 = (const float*)d_in[0];
    const float* volume2 = (const float*)d_in[1];
    const float* alpha   = (const float*)d_in[2];
    const float* dr      = (const float*)d_in[3];

    float* out_mults = (float*)d_out;                         // 32768 floats
    float* out_v2w   = out_mults + (size_t)NB * NF * NF;      // 33.5M floats
    float* out_grid  = out_v2w + (size_t)NB * NF * NPIX;      // 1.05M floats

    zero_mults_kernel<<<(NB * NF * NF) / 256, 256, 0, stream>>>(out_mults);

    warp_sample_kernel<<<(NB * NPIX) / 256, 256, 0, stream>>>(
        volume2, alpha, dr, out_v2w, out_grid);

    cross_gemm_kernel<<<NB * KCHUNKS, 512, 0, stream>>>(
        volume1, out_v2w, out_mults);
}